// TResLCBlock2d_76312978915840
// MI455X (gfx1250) — compile-verified
//
#include <hip/hip_runtime.h>

typedef __attribute__((ext_vector_type(2))) float v2f;
typedef __attribute__((ext_vector_type(8))) float v8f;

#define EPS_   1e-5f
#define SLOPE_ 0.01f
#define B_  16
#define C_  32
#define H_  64
#define W_  64
#define HW_ 4096            // H_*W_
#define HW9_ 36864          // HW_*9 : j-stride (elements) inside one cout's weight slab
#define KTOT_ 288           // C_*9
#define NELEM_ 2097152      // B_*C_*H_*W_
#define CNT_ 65536.0f       // B_*H_*W_  (BN reduction count)

__device__ __forceinline__ float leaky_(float v) { return v >= 0.0f ? v : SLOPE_ * v; }

// ---------------------------------------------------------------- zero sums
__global__ void zero_sums_kernel(float* __restrict__ sums) {
    sums[threadIdx.x] = 0.0f;   // 128 floats: sum1, sumsq1, sum2, sumsq2
}

// ---------------------------------------------------------------- BN stats -> packed {scale, shift}
__global__ void finalize_stats_kernel(const float* __restrict__ sum,
                                      const float* __restrict__ sumsq,
                                      const float* __restrict__ gamma,
                                      const float* __restrict__ beta,
                                      float* __restrict__ lslb) {   // lslb[2c]=scale, lslb[2c+1]=shift
    int c = threadIdx.x;                         // 32 threads
    float mean = sum[c] * (1.0f / CNT_);
    float var  = sumsq[c] * (1.0f / CNT_) - mean * mean;
    float rstd = rsqrtf(var + EPS_);
    float s = gamma[c] * rstd;
    lslb[2 * c]     = s;
    lslb[2 * c + 1] = beta[c] - mean * s;
}

// ---------------------------------------------------------------- Kernel 1: locally-connected GEMM
// Block = 64 threads (2 waves). Block -> one spatial position. Wave -> one 16-cout tile.
// WMMA f32 16x16x4:  D(16cout x 16batch) += A(16cout x 4K) * B(4K x 16batch)
// K labeling: K = tap*32 + j  (tap = 3x3 kernel index, j = cin). A 4-wide K group never
// crosses a tap boundary (4 | 32), so the tap is a compile-time, wave-uniform constant per
// iteration; the half-lane K shift (+2) is linear in j and folded into the base pointers.
__global__ void lc_wmma_kernel(const float* __restrict__ x,
                               const float* __restrict__ w,
                               const float* __restrict__ bias,
                               float* __restrict__ h1,
                               float* __restrict__ sum1,
                               float* __restrict__ sumsq1) {
    __builtin_assume(blockIdx.x < (unsigned)HW_);
    const int pos  = blockIdx.x;                 // 0..4095
    const int m    = pos >> 6, n = pos & 63;
    const int wave = threadIdx.x >> 5;           // 0..1 -> cout tile
    const int lane = threadIdx.x & 31;
    const int lid  = lane & 15;
    const int half = lane >> 4;
    const int i    = wave * 16 + lid;            // cout row this lane supplies for A

    const float* wbase = w + (i * C_ * HW_ + pos) * 9;   // slab for (cout=i, pos): + j*HW9_ + tap
    const float* wh    = wbase + half * (2 * HW9_);      // half-lane K shift folded in
    const float* xh    = x + lid * C_ * HW_ + half * (2 * HW_);

    // 9 tap base pointers + masks (constant-indexed after unroll -> registers)
    const float* xrp[9];
    float        pmsk[9];
#pragma unroll
    for (int t = 0; t < 9; ++t) {
        const int kk = t / 3, ll = t - 3 * (t / 3);
        int y = m + kk - 1, xc = n + ll - 1;
        bool inb = ((unsigned)y < (unsigned)H_) & ((unsigned)xc < (unsigned)W_);
        xrp[t]  = xh + (inb ? y : 0) * W_ + (inb ? xc : 0);
        pmsk[t] = inb ? 1.0f : 0.0f;
    }

    // pull the whole weight slab for this (cout, pos) into flight (in-bounds: unshifted base)
#pragma unroll
    for (int j = 0; j < C_; ++j)
        __builtin_prefetch(wbase + j * HW9_, 0, 0);

    v8f acc = {};
#pragma unroll
    for (int k0 = 0; k0 < KTOT_; k0 += 4) {
        const int r  = k0 >> 5;                  // tap 0..8   (compile-time constant)
        const int jb = k0 & 31;                  // cin base 0,4,...,28
        v2f a;                                   // immediate offsets from wh
        a.x = wh[jb * HW9_ + r];
        a.y = wh[jb * HW9_ + r + HW9_];
        const float msk = pmsk[r];
        v2f b;                                   // immediate offsets from tap pointer
        b.x = xrp[r][jb * HW_]       * msk;
        b.y = xrp[r][jb * HW_ + HW_] * msk;
        acc = __builtin_amdgcn_wmma_f32_16x16x4_f32(false, a, false, b,
                                                    (short)0, acc, false, false);
    }

    // D layout: VGPR v, lanes 0-15 -> M=v, lanes 16-31 -> M=v+8; N = lid = batch
    const int chb = wave * 16 + 8 * half;        // channel of acc[0] for this lane
    float*       hout = h1 + (lid * C_ + chb) * HW_ + pos;
    const float* bb   = bias + chb * HW_ + pos;
#pragma unroll
    for (int v = 0; v < 8; v++) {
        float val = acc[v] + bb[v * HW_];
        hout[v * HW_] = val;
        // BN1 partial sums: reduce across the 16 batch lanes of this half
        float s = val, q = val * val;
        s += __shfl_xor(s, 1);  q += __shfl_xor(q, 1);
        s += __shfl_xor(s, 2);  q += __shfl_xor(q, 2);
        s += __shfl_xor(s, 4);  q += __shfl_xor(q, 4);
        s += __shfl_xor(s, 8);  q += __shfl_xor(q, 8);
        if (lid == 0) {                          // lanes 0 and 16
            atomicAdd(&sum1[chb + v], s);
            atomicAdd(&sumsq1[chb + v], q);
        }
    }
}

// ---------------------------------------------------------------- Kernel 2: dense 3x3 conv (WMMA)
// Same K labeling. conv weight flat: (i*32+j)*9+tap = i*288 + j*9 + tap -> immediates.
__global__ void conv_wmma_kernel(const float* __restrict__ h1,
                                 const float* __restrict__ cw,
                                 const float* __restrict__ lslb1,  // packed {scale, shift} x32
                                 float* __restrict__ h2,
                                 float* __restrict__ sum2,
                                 float* __restrict__ sumsq2) {
    __builtin_assume(blockIdx.x < (unsigned)HW_);
    const int pos  = blockIdx.x;
    const int m    = pos >> 6, n = pos & 63;
    const int wave = threadIdx.x >> 5;
    const int lane = threadIdx.x & 31;
    const int lid  = lane & 15;
    const int half = lane >> 4;
    const int i    = wave * 16 + lid;

    const float* cwh = cw + i * KTOT_ + half * 18;        // + jb*9 + tap (immediate)
    const float* hb  = h1 + lid * C_ * HW_ + half * (2 * HW_);
    const float* lsh = lslb1 + half * 4;                  // + jb*2 (immediate), 8B-aligned

    const float* hrp[9];
    float        pmsk[9];
#pragma unroll
    for (int t = 0; t < 9; ++t) {
        const int kk = t / 3, ll = t - 3 * (t / 3);
        int y = m + kk - 1, xc = n + ll - 1;
        bool inb = ((unsigned)y < (unsigned)H_) & ((unsigned)xc < (unsigned)W_);
        hrp[t]  = hb + (inb ? y : 0) * W_ + (inb ? xc : 0);
        pmsk[t] = inb ? 1.0f : 0.0f;
    }

    v8f acc = {};
#pragma unroll
    for (int k0 = 0; k0 < KTOT_; k0 += 4) {
        const int r  = k0 >> 5;                  // tap (compile-time constant)
        const int jb = k0 & 31;                  // cin base
        v2f a;                                   // L2-resident weights, immediate offsets
        a.x = cwh[jb * 9 + r];
        a.y = cwh[jb * 9 + r + 9];
        // BN1 scale/shift for cin = jb+2*half (+1): packed b64 loads, cache hits
        v2f p0 = *(const v2f*)(lsh + jb * 2);
        v2f p1 = *(const v2f*)(lsh + jb * 2 + 2);
        const float msk = pmsk[r];
        float vX = hrp[r][jb * HW_];
        float vY = hrp[r][jb * HW_ + HW_];
        v2f b;                                   // BN1 + leaky, pad with 0 AFTER activation
        b.x = leaky_(vX * p0.x + p0.y) * msk;
        b.y = leaky_(vY * p1.x + p1.y) * msk;
        acc = __builtin_amdgcn_wmma_f32_16x16x4_f32(false, a, false, b,
                                                    (short)0, acc, false, false);
    }

    const int chb = wave * 16 + 8 * half;
    float* hout = h2 + (lid * C_ + chb) * HW_ + pos;
#pragma unroll
    for (int v = 0; v < 8; v++) {
        float val = acc[v];
        hout[v * HW_] = val;
        float s = val, q = val * val;
        s += __shfl_xor(s, 1);  q += __shfl_xor(q, 1);
        s += __shfl_xor(s, 2);  q += __shfl_xor(q, 2);
        s += __shfl_xor(s, 4);  q += __shfl_xor(q, 4);
        s += __shfl_xor(s, 8);  q += __shfl_xor(q, 8);
        if (lid == 0) {
            atomicAdd(&sum2[chb + v], s);
            atomicAdd(&sumsq2[chb + v], q);
        }
    }
}

// ---------------------------------------------------------------- Kernel 3: BN2 + residual + leaky
// float4: 4096 elems per channel, divisible by 4 -> channel uniform per vector
__global__ void final_kernel(const float4* __restrict__ h2,
                             const float4* __restrict__ x,
                             const float* __restrict__ lslb2,
                             float4* __restrict__ out) {
    int idx = blockIdx.x * blockDim.x + threadIdx.x;     // 0 .. NELEM_/4-1
    if (idx < NELEM_ / 4) {
        int c = (idx >> 10) & (C_ - 1);                  // (4*idx / 4096) % 32
        float s = lslb2[2 * c], t = lslb2[2 * c + 1];
        float4 h = h2[idx], xv = x[idx], o;
        o.x = leaky_(h.x * s + t + xv.x);
        o.y = leaky_(h.y * s + t + xv.y);
        o.z = leaky_(h.z * s + t + xv.z);
        o.w = leaky_(h.w * s + t + xv.w);
        out[idx] = o;
    }
}

// ---------------------------------------------------------------- launch
extern "C" void kernel_launch(void* const* d_in, const int* in_sizes, int n_in,
                              void* d_out, int out_size, void* d_ws, size_t ws_size,
                              hipStream_t stream) {
    (void)in_sizes; (void)n_in; (void)out_size; (void)ws_size;
    const float* x   = (const float*)d_in[0];
    const float* lcw = (const float*)d_in[1];
    const float* lcb = (const float*)d_in[2];
    const float* g1  = (const float*)d_in[3];
    const float* b1  = (const float*)d_in[4];
    const float* cw  = (const float*)d_in[5];
    const float* g2  = (const float*)d_in[6];
    const float* b2  = (const float*)d_in[7];
    float* out = (float*)d_out;

    float* ws = (float*)d_ws;
    float* h1     = ws;                    // 2,097,152 floats
    float* h2     = ws + NELEM_;           // 2,097,152 floats
    float* sum1   = ws + 2 * NELEM_;       // 32
    float* sumsq1 = sum1 + 32;
    float* sum2   = sumsq1 + 32;
    float* sumsq2 = sum2 + 32;
    float* lslb1  = sumsq2 + 32;           // 64 (packed {scale, shift})
    float* lslb2  = lslb1 + 64;            // 64

    zero_sums_kernel<<<1, 128, 0, stream>>>(sum1);   // zeros sum1..sumsq2 (contiguous 128 floats)
    lc_wmma_kernel<<<HW_, 64, 0, stream>>>(x, lcw, lcb, h1, sum1, sumsq1);
    finalize_stats_kernel<<<1, 32, 0, stream>>>(sum1, sumsq1, g1, b1, lslb1);
    conv_wmma_kernel<<<HW_, 64, 0, stream>>>(h1, cw, lslb1, h2, sum2, sumsq2);
    finalize_stats_kernel<<<1, 32, 0, stream>>>(sum2, sumsq2, g2, b2, lslb2);
    final_kernel<<<(NELEM_ / 4 + 255) / 256, 256, 0, stream>>>(
        (const float4*)h2, (const float4*)x, lslb2, (float4*)out);
}